// MeshPool_83820581748889
// MI455X (gfx1250) — compile-verified
//
#include <hip/hip_runtime.h>

// pooled = (rel / rowsum(rel)) @ feat
//   rel  : [12000, 24000] f32  (streamed once from HBM; 1.152 GB -> ~50us floor @ 23.3 TB/s)
//   feat : [24000, 32]    f32  (3 MB, cache-resident)
//   out  : [12000, 32]    f32
//
// Strategy: V_WMMA_F32_16X16X4_F32 tiles of 16 rows x 16 cols, K stepped by 4.
// K split 8 ways for occupancy (6000 waves); deterministic per-segment partial
// sums in d_ws, reduced + normalized by a second kernel.

typedef __attribute__((ext_vector_type(2))) float v2f;
typedef __attribute__((ext_vector_type(8))) float v8f;

namespace {
constexpr int M = 12000;          // EDGE_TARGET (output rows)
constexpr int K = 24000;          // NUM_EDGES   (reduction dim)
constexpr int N = 32;             // FEAT_DIM
constexpr int KSPLIT = 8;         // K segments (one per wave in a block)
constexpr int KSEG = K / KSPLIT;  // 3000
constexpr int TILES = M / 16;     // 750 M-tiles
}

__global__ __launch_bounds__(256) void meshpool_wmma_partial(
    const float* __restrict__ feat, const float* __restrict__ rel,
    float* __restrict__ pOut, float* __restrict__ pRs)
{
    const int lane = threadIdx.x & 31;
    const int seg  = threadIdx.x >> 5;   // wave id in block = K segment 0..7
    const int tile = blockIdx.x;         // M-tile 0..749
    const int col  = lane & 15;
    const int hi   = lane >> 4;          // 0: K%4 in {0,1}; 1: K%4 in {2,3}

    const int m0 = tile * 16;
    const int k0 = seg * KSEG;

    // A fragment (16x4 f32): lane holds row (m0+col), K offsets 2*hi + {0,1}
    const float* ap = rel + (size_t)(m0 + col) * K + k0 + 2 * hi;
    // B fragment (4x16 f32): lane holds col, K offsets 2*hi + {0,1}
    const float* bp = feat + (size_t)(k0 + 2 * hi) * N + col;

    v8f c0 = {};          // cols 0..15
    v8f c1 = {};          // cols 16..31
    float rs = 0.0f;      // partial row sum (this lane's K phases)

#pragma unroll 4
    for (int i = 0; i < KSEG / 4; ++i) {
        v2f a = *(const v2f*)ap;               // global_load_b64, 100% line reuse
        float b00 = bp[0];
        float b01 = bp[N];
        float b10 = bp[16];
        float b11 = bp[16 + N];
        ap += 4;
        bp += 4 * N;
        rs += a.x + a.y;
        v2f b0; b0.x = b00; b0.y = b01;
        v2f b1; b1.x = b10; b1.y = b11;
        c0 = __builtin_amdgcn_wmma_f32_16x16x4_f32(false, a, false, b0,
                                                   (short)0, c0, false, false);
        c1 = __builtin_amdgcn_wmma_f32_16x16x4_f32(false, a, false, b1,
                                                   (short)0, c1, false, false);
    }

    // Lanes L and L^16 cover the same row with complementary K phases.
    rs += __shfl_xor(rs, 16, 32);   // now every lane holds seg-rowsum(row = col)

    // Store unnormalized partials: D layout = VGPR j -> row j + 8*hi, col = lane&15.
    float* po = pOut + (size_t)seg * (M * N) + (size_t)m0 * N + col;
#pragma unroll
    for (int j = 0; j < 8; ++j) {
        const int r = j + 8 * hi;
        po[(size_t)r * N]      = c0[j];
        po[(size_t)r * N + 16] = c1[j];
    }
    if (hi == 0)
        pRs[(size_t)seg * M + m0 + col] = rs;
}

__global__ __launch_bounds__(256) void meshpool_reduce(
    const float* __restrict__ pOut, const float* __restrict__ pRs,
    float* __restrict__ out)
{
    const int idx = blockIdx.x * 256 + threadIdx.x;
    if (idx >= M * N) return;
    const int row = idx >> 5;   // idx / 32
    float acc = 0.0f, rsum = 0.0f;
#pragma unroll
    for (int s = 0; s < KSPLIT; ++s) {
        acc  += pOut[(size_t)s * (M * N) + idx];
        rsum += pRs[(size_t)s * M + row];
    }
    out[idx] = acc / rsum;      // (rel @ feat) / rowsum == (rel/rowsum) @ feat
}

extern "C" void kernel_launch(void* const* d_in, const int* in_sizes, int n_in,
                              void* d_out, int out_size, void* d_ws, size_t ws_size,
                              hipStream_t stream)
{
    const float* feat = (const float*)d_in[0];   // [24000, 32]
    const float* rel  = (const float*)d_in[1];   // [12000, 24000]
    float* out = (float*)d_out;                  // [12000, 32]

    // Workspace: KSPLIT*M*N partial outputs + KSPLIT*M partial row sums (~12.7 MB).
    float* pOut = (float*)d_ws;
    float* pRs  = pOut + (size_t)KSPLIT * M * N;

    meshpool_wmma_partial<<<TILES, 256, 0, stream>>>(feat, rel, pOut, pRs);
    meshpool_reduce<<<(M * N + 255) / 256, 256, 0, stream>>>(pOut, pRs, out);
}